// GraphConv_44298292691098
// MI455X (gfx1250) — compile-verified
//
#include <hip/hip_runtime.h>

typedef __attribute__((ext_vector_type(2))) float    v2f;
typedef __attribute__((ext_vector_type(8))) float    v8f;
typedef __attribute__((ext_vector_type(4))) unsigned v4u;
typedef __attribute__((ext_vector_type(4))) int      v4i;
typedef __attribute__((ext_vector_type(8))) int      v8i;

#define N_NODES   100000
#define N_EDGES   1600000
#define IN_FEATS  512
#define OUT_FEATS 128
#define ROW_TILES ((N_NODES + 15) / 16)   // 6250, exact

// ---------------------------------------------------------------- utilities

__global__ void zero_kernel(float* __restrict__ p, size_t n) {
    size_t i = (size_t)blockIdx.x * blockDim.x + threadIdx.x;
    size_t stride = (size_t)gridDim.x * blockDim.x;
    for (; i < n; i += stride) p[i] = 0.0f;
}

__global__ void degree_kernel(const int* __restrict__ src,
                              const int* __restrict__ dst,
                              float* __restrict__ deg_out,
                              float* __restrict__ deg_in) {
    int e = blockIdx.x * blockDim.x + threadIdx.x;
    if (e < N_EDGES) {
        atomicAdd(&deg_out[src[e]], 1.0f);
        atomicAdd(&deg_in[dst[e]], 1.0f);
    }
}

// ---------------------------------------------------------------- GEMM
// h[N_NODES x 128] = (feat * rsqrt(out_deg)) @ W[512 x 128]
// 8 waves / block; whole W staged in LDS (256 KB, legal on 320 KB WGP)
// via one Tensor Data Mover DMA (TENSORcnt), then fp32 WMMA 16x16x4.

__global__ __launch_bounds__(256) void gemm_kernel(
        const float* __restrict__ feat,
        const float* __restrict__ weight,
        const float* __restrict__ deg_out,
        float* __restrict__ h) {
    extern __shared__ float ldsW[];   // IN_FEATS * OUT_FEATS floats = 256 KB

#if __has_builtin(__builtin_amdgcn_tensor_load_to_lds)
    // ---- TDM stage: one DMA of the whole 512x128 fp32 weight into LDS ----
    if (threadIdx.x < 32) {           // one wave issues (EXEC ignored by TDM,
                                      // so the other waves must branch around)
        const unsigned lds_off = (unsigned)(unsigned long long)(const void*)ldsW;
        const unsigned long long ga = (unsigned long long)weight;
        // D# group 0: count=1 | lds_addr | global_addr[56:0] | type=2
        v4u g0 = { 1u,
                   lds_off,
                   (unsigned)ga,
                   (unsigned)((ga >> 32) & 0x01FFFFFFu) | (2u << 30) };
        // D# group 1: data_size=2 (4B); tensor_dim0=128, tensor_dim1=512;
        // tile_dim0=128, tile_dim1=512, tile_dim2=0; tensor_dim0_stride=128
        v8i g1 = { (int)(2u << 16),          // workgroup_mask=0, data_size=4B
                   (int)(128u << 16),        // tensor_dim0[15:0]=128
                   (int)(512u << 16),        // dim0 hi=0, tensor_dim1[15:0]=512
                   (int)(128u << 16),        // dim1 hi=0, tile_dim0=128
                   512,                      // tile_dim1=512, tile_dim2=0
                   128,                      // tensor_dim0_stride[31:0]=128
                   0, 0 };                   // stride hi, tensor_dim1_stride
        v4i g2 = { 0, 0, 0, 0 };             // 2D tensor: groups 2/3 unused
        v4i g3 = { 0, 0, 0, 0 };
#if defined(__clang_major__) && (__clang_major__ >= 23)
        v8i g4 = { 0, 0, 0, 0, 0, 0, 0, 0 };
        __builtin_amdgcn_tensor_load_to_lds(g0, g1, g2, g3, g4, 0);
#else
        __builtin_amdgcn_tensor_load_to_lds(g0, g1, g2, g3, 0);
#endif
        __builtin_amdgcn_s_wait_tensorcnt(0);
    }
#else
    // ---- fallback: cooperative float4 load ----
    {
        const float4* wsrc = (const float4*)weight;
        float4* wdst = (float4*)ldsW;
        const int total4 = IN_FEATS * OUT_FEATS / 4;
        for (int i = threadIdx.x; i < total4; i += 256) wdst[i] = wsrc[i];
    }
#endif
    __syncthreads();

    const int wave = threadIdx.x >> 5;
    const int lane = threadIdx.x & 31;
    const int rowTile = blockIdx.x * 8 + wave;
    if (rowTile >= ROW_TILES) return;
    const int row0 = rowTile * 16;

    // A-fragment addressing per ISA 16x4 f32 layout:
    // lanes 0-15 hold (M=lane, K=0/1), lanes 16-31 hold (M=lane-16, K=2/3)
    const int m  = lane & 15;
    const int kb = (lane >> 4) * 2;
    const int n  = lane & 15;            // B/D column within 16-wide tile
    const int row = row0 + m;

    const float* arow = feat + (size_t)row * IN_FEATS + kb;
    float d = deg_out[row];
    d = d < 1.0f ? 1.0f : d;
    const float scale = rsqrtf(d);       // left-normalization folded into A

    v8f acc[8] = {};                     // 8 N-tiles x (16x16 f32) accumulators

    for (int k0 = 0; k0 < IN_FEATS; k0 += 4) {
        v2f a;
        a.x = arow[k0]     * scale;
        a.y = arow[k0 + 1] * scale;
        const float* wrow0 = ldsW + (size_t)(k0 + kb) * OUT_FEATS;
        const float* wrow1 = wrow0 + OUT_FEATS;
#pragma unroll
        for (int t = 0; t < 8; ++t) {
            v2f b;
            b.x = wrow0[t * 16 + n];
            b.y = wrow1[t * 16 + n];
            acc[t] = __builtin_amdgcn_wmma_f32_16x16x4_f32(
                false, a, false, b, (short)0, acc[t], false, false);
        }
    }

    // D layout: VGPR r -> lanes 0-15: (M=r, N=lane); lanes 16-31: (M=r+8, N=lane-16)
    const int mbase = (lane < 16) ? 0 : 8;
#pragma unroll
    for (int t = 0; t < 8; ++t) {
        float* cptr = h + (size_t)(row0 + mbase) * OUT_FEATS + t * 16 + n;
#pragma unroll
        for (int r = 0; r < 8; ++r) {
            cptr[(size_t)r * OUT_FEATS] = acc[t][r];
        }
    }
}

// ---------------------------------------------------------------- edges
// One wave per edge; lane handles 4 consecutive output channels.
// h and out are L2-resident (51 MB each vs 192 MB L2) -> L2-side atomics.

__global__ __launch_bounds__(256) void edge_kernel(
        const float* __restrict__ h,
        const float* __restrict__ ew,
        const int* __restrict__ src,
        const int* __restrict__ dst,
        float* __restrict__ out) {
    const int e = blockIdx.x * 8 + (threadIdx.x >> 5);
    if (e >= N_EDGES) return;
    const int lane = threadIdx.x & 31;
    const int s = src[e];
    const int t = dst[e];
    const float w = ew[e];
    const float4 v = ((const float4*)(h + (size_t)s * OUT_FEATS))[lane];
    float* op = out + (size_t)t * OUT_FEATS + lane * 4;
    atomicAdd(op + 0, v.x * w);
    atomicAdd(op + 1, v.y * w);
    atomicAdd(op + 2, v.z * w);
    atomicAdd(op + 3, v.w * w);
}

// ---------------------------------------------------------------- finalize

__global__ void finalize_kernel(float* __restrict__ out,
                                const float* __restrict__ deg_in,
                                const float* __restrict__ bias) {
    size_t i = (size_t)blockIdx.x * blockDim.x + threadIdx.x;
    if (i >= (size_t)N_NODES * OUT_FEATS) return;
    const int node = (int)(i >> 7);
    const int c = (int)(i & 127);
    float d = deg_in[node];
    d = d < 1.0f ? 1.0f : d;
    out[i] = out[i] * rsqrtf(d) + bias[c];
}

// ---------------------------------------------------------------- launch

extern "C" void kernel_launch(void* const* d_in, const int* in_sizes, int n_in,
                              void* d_out, int out_size, void* d_ws, size_t ws_size,
                              hipStream_t stream) {
    (void)in_sizes; (void)n_in; (void)out_size; (void)ws_size;

    const float* feat   = (const float*)d_in[0];
    const float* weight = (const float*)d_in[1];
    const float* bias   = (const float*)d_in[2];
    const float* ew     = (const float*)d_in[3];
    const int*   src    = (const int*)d_in[4];
    const int*   dst    = (const int*)d_in[5];
    float* out = (float*)d_out;

    // workspace: [deg_out N][deg_in N][h N*128]
    float* deg_out = (float*)d_ws;
    float* deg_in  = deg_out + N_NODES;
    float* h       = deg_in + N_NODES;

    // zero degree arrays + output accumulator
    zero_kernel<<<1024, 256, 0, stream>>>(deg_out, (size_t)2 * N_NODES);
    zero_kernel<<<4096, 256, 0, stream>>>(out, (size_t)N_NODES * OUT_FEATS);

    // degrees
    degree_kernel<<<(N_EDGES + 255) / 256, 256, 0, stream>>>(src, dst, deg_out, deg_in);

    // projection GEMM with fp32 WMMA; 256 KB dynamic LDS for W (TDM-staged)
    const int gemm_blocks = (ROW_TILES + 7) / 8;   // 782
    gemm_kernel<<<gemm_blocks, 256, IN_FEATS * OUT_FEATS * sizeof(float), stream>>>(
        feat, weight, deg_out, h);

    // edge scatter-add into out
    edge_kernel<<<(N_EDGES + 7) / 8, 256, 0, stream>>>(h, ew, src, dst, out);

    // right-normalize + bias
    const size_t total = (size_t)N_NODES * OUT_FEATS;
    finalize_kernel<<<(unsigned)((total + 255) / 256), 256, 0, stream>>>(out, deg_in, bias);
}